// GraphSAGE_40862318854556
// MI455X (gfx1250) — compile-verified
//
#include <hip/hip_runtime.h>

// GraphSAGE 2-layer, fp32, MI455X (gfx1250, wave32).
// Edge gather (~0.7 GB HBM traffic) is the roofline bottleneck; GEMMs
// (6.5 GFLOP) run on the fp32 WMMA path (V_WMMA_F32_16X16X4_F32) with the
// weight operand async-staged into LDS (global_load_async_to_lds_b128).

#define N0 360448
#define N1 22528
#define N2 2048
#define E0 337920
#define E1 20480
#define D  256

#define BK 128          // K-chunk staged in LDS per stage
#define LDS_STRIDE 132  // floats per staged weight row (128 + 4 pad -> conflict-free)
#define COLS 64         // output columns (weight rows) per block

typedef __attribute__((ext_vector_type(2))) float v2f;
typedef __attribute__((ext_vector_type(8))) float v8f;

// -------------------------------------------------- async global -> LDS copy
__device__ __forceinline__ void async_ld_b128(unsigned lds_addr, const void* gaddr) {
    asm volatile("global_load_async_to_lds_b128 %0, %1, off"
                 :: "v"(lds_addr), "v"(gaddr)
                 : "memory");
}
__device__ __forceinline__ void wait_async0() {
    asm volatile("s_wait_asynccnt 0x0" ::: "memory");
}

// ---------------------------------------------------------------- zero scratch
__global__ void sage_zero(float* __restrict__ p, long long n) {
    long long i = (long long)blockIdx.x * blockDim.x + threadIdx.x;
    long long stride = (long long)gridDim.x * blockDim.x;
    for (; i < n; i += stride) p[i] = 0.0f;
}

// ------------------------------------------------------- edge scatter (mean #)
// 4 edges per 256-thread block; 64 lanes per edge, float4 per lane.
__global__ void sage_scatter(const float* __restrict__ xsrc,
                             const int* __restrict__ src,
                             const int* __restrict__ dst,
                             float* __restrict__ sum,
                             float* __restrict__ cnt,
                             int nedges) {
    int e = blockIdx.x * 4 + (threadIdx.x >> 6);
    int t = threadIdx.x & 63;
    if (e >= nedges) return;
    int s = src[e];
    int d = dst[e];
    const float4* __restrict__ xr = (const float4*)(xsrc + (size_t)s * D);
    float* __restrict__ out = sum + (size_t)d * D;
    float4 v = xr[t];
    atomicAdd(out + t * 4 + 0, v.x);
    atomicAdd(out + t * 4 + 1, v.y);
    atomicAdd(out + t * 4 + 2, v.z);
    atomicAdd(out + t * 4 + 3, v.w);
    if (t == 0) atomicAdd(cnt + d, 1.0f);
}

// ------------------------------------------------------------ fused SAGE GEMM
// Block = 128 threads = 4 waves; block computes a 64x64 output tile.
// Wave w owns rows [blockIdx.y*64 + 16w, +16) and all 64 columns:
// 4 accumulators (16x16 tiles), A float2 reused across 4 WMMAs.
// Weight operand staged in LDS in 4 stages (Wl lo/hi, Wr lo/hi) via
// global_load_async_to_lds_b128 + s_wait_asynccnt + barrier.
//
// Operand layout (ISA 7.12.2, wave32), per lane L (nl=L&15, kh=L>>4):
//   A 16x4 f32:  m = nl ; VGPR j = A[m][2*kh + j]
//   B  4x16 f32: n = nl ; VGPR j = B[2*kh + j][n] = W[n][k]
//   C/D 16x16:   n = nl ; VGPR v -> M = v + 8*kh
template <int RELU>
__global__ void sage_wmma_gemm(const float* __restrict__ sum,
                               const float* __restrict__ cnt,
                               const float* __restrict__ xt,
                               const float* __restrict__ Wl,
                               const float* __restrict__ bl,
                               const float* __restrict__ Wr,
                               float* __restrict__ out) {
    extern __shared__ float smem[];      // COLS * LDS_STRIDE floats (33 KB)
    const int tid  = threadIdx.x;        // 0..127
    const int wv   = tid >> 5;
    const int lane = tid & 31;
    const int nl   = lane & 15;
    const int kh   = lane >> 4;

    const int colBase = blockIdx.x * COLS;
    const int mTile   = blockIdx.y * 64 + wv * 16;
    const int row     = mTile + nl;      // A-operand row for this lane

    const float scale = 1.0f / fmaxf(cnt[row], 1.0f);
    const unsigned ldsBase = (unsigned)(unsigned long long)smem;

    v8f acc[4] = {};

    for (int stage = 0; stage < 4; ++stage) {
        const float* __restrict__ W    = (stage < 2) ? Wl  : Wr;
        const float* __restrict__ Asrc = (stage < 2) ? sum : xt;
        const float  asc  = (stage < 2) ? scale : 1.0f;
        const int    kOff = (stage & 1) * BK;

        __syncthreads();  // previous stage's LDS readers are done
        // stage W[colBase..colBase+63][kOff..kOff+127] -> LDS (2048 x b128)
#pragma unroll
        for (int i = 0; i < 16; ++i) {
            int id = i * 128 + tid;
            int r  = id >> 5;            // 32 b128 chunks per 128-float row
            int c  = id & 31;
            const float* g = W + (size_t)(colBase + r) * D + kOff + c * 4;
            async_ld_b128(ldsBase + (unsigned)(r * LDS_STRIDE + c * 4) * 4u, g);
        }
        wait_async0();
        __syncthreads();  // staged weights visible to all waves

        const float2* __restrict__ ap =
            (const float2*)(Asrc + (size_t)row * D + kOff) + kh;
#pragma unroll 4
        for (int kk = 0; kk < BK / 4; ++kk) {
            float2 av = ap[kk * 2];
            v2f a; a[0] = av.x * asc; a[1] = av.y * asc;
#pragma unroll
            for (int t = 0; t < 4; ++t) {
                const float2 bv = *(const float2*)
                    (smem + (t * 16 + nl) * LDS_STRIDE + kk * 4 + 2 * kh);
                v2f b; b[0] = bv.x; b[1] = bv.y;
                acc[t] = __builtin_amdgcn_wmma_f32_16x16x4_f32(
                             false, a, false, b, (short)0, acc[t], false, false);
            }
        }
    }

    // epilogue: bias (+ReLU), 4 column tiles x 8 rows
    const int mbase = mTile + kh * 8;
#pragma unroll
    for (int t = 0; t < 4; ++t) {
        const int   col  = colBase + t * 16 + nl;
        const float bias = bl[col];
#pragma unroll
        for (int v = 0; v < 8; ++v) {
            float val = acc[t][v] + bias;
            if (RELU) val = fmaxf(val, 0.0f);
            out[(size_t)(mbase + v) * D + col] = val;
        }
    }
}

// ---------------------------------------------------------------------- launch
extern "C" void kernel_launch(void* const* d_in, const int* in_sizes, int n_in,
                              void* d_out, int out_size, void* d_ws, size_t ws_size,
                              hipStream_t stream) {
    const float* x    = (const float*)d_in[0];
    const int*   src0 = (const int*)d_in[1];
    const int*   dst0 = (const int*)d_in[2];
    const int*   src1 = (const int*)d_in[3];
    const int*   dst1 = (const int*)d_in[4];
    const float* Wl0  = (const float*)d_in[5];
    const float* bl0  = (const float*)d_in[6];
    const float* Wr0  = (const float*)d_in[7];
    const float* Wl1  = (const float*)d_in[8];
    const float* bl1  = (const float*)d_in[9];
    const float* Wr1  = (const float*)d_in[10];
    float* out = (float*)d_out;

    // workspace layout (floats): [sum0 | cnt0 | sum1 | cnt1 | h]
    float* ws   = (float*)d_ws;
    float* sum0 = ws;
    float* cnt0 = sum0 + (size_t)N1 * D;
    float* sum1 = cnt0 + N1;
    float* cnt1 = sum1 + (size_t)N2 * D;
    float* h    = cnt1 + N2;

    const long long nzero = (long long)N1 * D + N1 + (long long)N2 * D + N2;
    sage_zero<<<4096, 256, 0, stream>>>(ws, nzero);

    const size_t ldsBytes = (size_t)COLS * LDS_STRIDE * sizeof(float);  // 33792

    // layer 0: aggregate E0 edges, fused WMMA GEMM + ReLU -> h [N1,D]
    sage_scatter<<<E0 / 4, 256, 0, stream>>>(x, src0, dst0, sum0, cnt0, E0);
    {
        dim3 grid(D / COLS, N1 / 64);
        sage_wmma_gemm<1><<<grid, 128, ldsBytes, stream>>>(sum0, cnt0, x,
                                                           Wl0, bl0, Wr0, h);
    }

    // layer 1: aggregate E1 edges, fused WMMA GEMM -> out [N2,D]
    sage_scatter<<<E1 / 4, 256, 0, stream>>>(h, src1, dst1, sum1, cnt1, E1);
    {
        dim3 grid(D / COLS, N2 / 64);
        sage_wmma_gemm<0><<<grid, 128, ldsBytes, stream>>>(sum1, cnt1, h,
                                                           Wl1, bl1, Wr1, out);
    }
}